// BotRGCN_4layers_32495722562040
// MI455X (gfx1250) — compile-verified
//
#include <hip/hip_runtime.h>

typedef __attribute__((ext_vector_type(2))) float v2f;
typedef __attribute__((ext_vector_type(8))) float v8f;

#define DD 128
#define RR 5

__device__ __forceinline__ float lrelu_f(float x) { return x > 0.f ? x : 0.01f * x; }

// ---------------------------------------------------------------------------
// One-time 128x128 transpose: Wt[j][k] = W[k][j]  (so B-frags are contiguous)
// ---------------------------------------------------------------------------
__global__ void transpose128_kernel(const float* __restrict__ W, float* __restrict__ Wt) {
  int i = blockIdx.x * blockDim.x + threadIdx.x;   // 0..16383
  int k = i >> 7, j = i & 127;
  Wt[(size_t)j * DD + k] = W[(size_t)k * DD + j];
}

// ---------------------------------------------------------------------------
// Y[N x 128] = act(X[N x 128] @ W[128 x 128] + bias), W given TRANSPOSED (Wt[n][k]).
// 128 threads = 4 waves; block handles 32 rows; wave w handles cols [32w,32w+32).
// 2x2 register tiling: each wave owns 2 row-tiles x 2 col-tiles -> 4 WMMAs per
// k-chunk against 2 LDS A-loads + 2 global b64 B-loads.
// A-frag (16x4 f32): lane<16 -> K=k,k+1 ; lane>=16 -> K=k+2,k+3 (row M = lane&15)
// B-frag (4x16 f32): same K split, col N = lane&15
// C/D (16x16 f32):   VGPR v -> row v (lane<16) / v+8 (lane>=16), col = lane&15
// Tail rows (N % 32) zero-padded in LDS; stores guarded after all WMMAs.
// ---------------------------------------------------------------------------
__global__ void __launch_bounds__(128)
gemm128_kernel(const float* __restrict__ X, const float* __restrict__ Wt,
               const float* __restrict__ bias, float* __restrict__ Y,
               int N, int act) {
  __shared__ float xs[32 * DD];
  const int row0 = blockIdx.x * 32;
  const int tid  = threadIdx.x;

  // cooperative stage of 32x128 x-tile into LDS, zero-padded past N
  for (int i = tid * 4; i < 32 * DD; i += 128 * 4) {
    const int gr = row0 + (i >> 7);
    float4 v = make_float4(0.f, 0.f, 0.f, 0.f);
    if (gr < N) v = *(const float4*)(X + (size_t)gr * DD + (i & 127));
    *(float4*)(xs + i) = v;
  }
  __syncthreads();

  const int wave = tid >> 5;
  const int lane = tid & 31;
  const int m    = lane & 15;
  const int koff = (lane >> 4) << 1;   // 0 or 2
  const int n    = lane & 15;
  const int n0   = wave * 32;
  const int n1   = n0 + 16;

  const float* WtB0 = Wt + (size_t)(n0 + n) * DD;  // col (n0+n), contiguous in k
  const float* WtB1 = Wt + (size_t)(n1 + n) * DD;

  v8f c00 = {}, c01 = {}, c10 = {}, c11 = {};
  for (int k = 0; k < DD; k += 4) {
    v2f a0, a1, b0, b1;
    a0.x = xs[m * DD + k + koff];
    a0.y = xs[m * DD + k + koff + 1];
    a1.x = xs[(m + 16) * DD + k + koff];
    a1.y = xs[(m + 16) * DD + k + koff + 1];
    b0 = *(const v2f*)(WtB0 + k + koff);           // 8B-aligned b64 load
    b1 = *(const v2f*)(WtB1 + k + koff);
    c00 = __builtin_amdgcn_wmma_f32_16x16x4_f32(false, a0, false, b0, (short)0, c00, false, false);
    c01 = __builtin_amdgcn_wmma_f32_16x16x4_f32(false, a0, false, b1, (short)0, c01, false, false);
    c10 = __builtin_amdgcn_wmma_f32_16x16x4_f32(false, a1, false, b0, (short)0, c10, false, false);
    c11 = __builtin_amdgcn_wmma_f32_16x16x4_f32(false, a1, false, b1, (short)0, c11, false, false);
  }

  const float bv0 = bias ? bias[n0 + n] : 0.f;
  const float bv1 = bias ? bias[n1 + n] : 0.f;
  const int r0 = row0 + ((lane >> 4) << 3);        // row-tile 0
  const int r1 = r0 + 16;                          // row-tile 1
#pragma unroll
  for (int v = 0; v < 8; ++v) {
    if (r0 + v < N) {
      float y0 = c00[v] + bv0, y1 = c01[v] + bv1;
      if (act) { y0 = lrelu_f(y0); y1 = lrelu_f(y1); }
      Y[(size_t)(r0 + v) * DD + n0 + n] = y0;
      Y[(size_t)(r0 + v) * DD + n1 + n] = y1;
    }
    if (r1 + v < N) {
      float y0 = c10[v] + bv0, y1 = c11[v] + bv1;
      if (act) { y0 = lrelu_f(y0); y1 = lrelu_f(y1); }
      Y[(size_t)(r1 + v) * DD + n0 + n] = y0;
      Y[(size_t)(r1 + v) * DD + n1 + n] = y1;
    }
  }
}

// ---------------------------------------------------------------------------
// Y[:, colofs:colofs+32] = lrelu(X[N x K] @ W[K x 32] + bias[32])
// 256 threads = 8 waves; each wave owns 16 rows, computes both 16-col tiles.
// K padded to multiple of 4 with masked (zero) loads.
// ---------------------------------------------------------------------------
__global__ void __launch_bounds__(256)
proj_kernel(const float* __restrict__ X, int K, const float* __restrict__ W,
            const float* __restrict__ bias, float* __restrict__ Y,
            int colofs, int N) {
  const int wave = threadIdx.x >> 5;
  const int lane = threadIdx.x & 31;
  const int row0 = (blockIdx.x * 8 + wave) * 16;
  if (row0 >= N) return;   // whole-wave uniform exit; EXEC stays all-ones for WMMA

  const int m    = lane & 15;
  const int koff = (lane >> 4) << 1;
  const int n    = lane & 15;

  v8f c0 = {}; v8f c1 = {};
  const int Kp = (K + 3) & ~3;
  for (int k = 0; k < Kp; k += 4) {
    const int k0 = k + koff, k1 = k + koff + 1;
    v2f a, b0, b1;
    a.x  = (k0 < K) ? X[(size_t)(row0 + m) * K + k0] : 0.f;
    a.y  = (k1 < K) ? X[(size_t)(row0 + m) * K + k1] : 0.f;
    b0.x = (k0 < K) ? W[(size_t)k0 * 32 + n] : 0.f;
    b0.y = (k1 < K) ? W[(size_t)k1 * 32 + n] : 0.f;
    b1.x = (k0 < K) ? W[(size_t)k0 * 32 + 16 + n] : 0.f;
    b1.y = (k1 < K) ? W[(size_t)k1 * 32 + 16 + n] : 0.f;
    c0 = __builtin_amdgcn_wmma_f32_16x16x4_f32(false, a, false, b0, (short)0, c0, false, false);
    c1 = __builtin_amdgcn_wmma_f32_16x16x4_f32(false, a, false, b1, (short)0, c1, false, false);
  }

  const float bv0 = bias[n];
  const float bv1 = bias[16 + n];
  const int rbase = row0 + ((lane >> 4) << 3);
#pragma unroll
  for (int v = 0; v < 8; ++v) {
    Y[(size_t)(rbase + v) * DD + colofs + n]      = lrelu_f(c0[v] + bv0);
    Y[(size_t)(rbase + v) * DD + colofs + 16 + n] = lrelu_f(c1[v] + bv1);
  }
}

// ---------------------------------------------------------------------------
// Edge-count / mean-normalization helpers
// ---------------------------------------------------------------------------
__global__ void zero_kernel(float* __restrict__ p, int n) {
  int i = blockIdx.x * blockDim.x + threadIdx.x;
  if (i < n) p[i] = 0.f;
}

__global__ void count_kernel(const int* __restrict__ ei, const int* __restrict__ et,
                             float* __restrict__ cnt, int E) {
  int e = blockIdx.x * blockDim.x + threadIdx.x;
  if (e < E) atomicAdd(&cnt[(size_t)ei[E + e] * RR + et[e]], 1.0f);
}

__global__ void invert_kernel(float* __restrict__ cnt, int n) {
  int i = blockIdx.x * blockDim.x + threadIdx.x;
  if (i < n) cnt[i] = 1.0f / fmaxf(cnt[i], 1.0f);
}

// ---------------------------------------------------------------------------
// Per-edge scatter: Y[dst] += h[etype][src] * inv_cnt[dst,etype]
// One wave per edge; float4 load per lane, 4 fp32 global atomics per lane.
// ---------------------------------------------------------------------------
__global__ void __launch_bounds__(256)
scatter_kernel(const int* __restrict__ ei, const int* __restrict__ et,
               const float* __restrict__ h, const float* __restrict__ inv,
               float* __restrict__ Y, int E, int N) {
  const int e = blockIdx.x * 8 + (threadIdx.x >> 5);
  if (e >= E) return;
  const int lane = threadIdx.x & 31;
  const int src = ei[e];
  const int dst = ei[E + e];
  const int t   = et[e];
  const float w = inv[(size_t)dst * RR + t];
  const float4 v = *(const float4*)(h + ((size_t)t * N + src) * DD + lane * 4);
  float* yd = Y + (size_t)dst * DD + lane * 4;
  atomicAdd(yd + 0, v.x * w);
  atomicAdd(yd + 1, v.y * w);
  atomicAdd(yd + 2, v.z * w);
  atomicAdd(yd + 3, v.w * w);
}

// ---------------------------------------------------------------------------
// out[n, 0:2] = X[n] @ W_o2[128 x 2] + b_o2
// ---------------------------------------------------------------------------
__global__ void out_kernel(const float* __restrict__ X, const float* __restrict__ W,
                           const float* __restrict__ b, float* __restrict__ out, int N) {
  int n = blockIdx.x * blockDim.x + threadIdx.x;
  if (n >= N) return;
  float s0 = b[0], s1 = b[1];
  const float* x = X + (size_t)n * DD;
#pragma unroll 8
  for (int d = 0; d < DD; ++d) {
    float xv = x[d];
    s0 += xv * W[d * 2 + 0];
    s1 += xv * W[d * 2 + 1];
  }
  out[(size_t)n * 2 + 0] = s0;
  out[(size_t)n * 2 + 1] = s1;
}

// ---------------------------------------------------------------------------
extern "C" void kernel_launch(void* const* d_in, const int* in_sizes, int n_in,
                              void* d_out, int out_size, void* d_ws, size_t ws_size,
                              hipStream_t stream) {
  const float* des      = (const float*)d_in[0];
  const float* tweet    = (const float*)d_in[1];
  const float* num_prop = (const float*)d_in[2];
  const float* cat_prop = (const float*)d_in[3];
  const int*   ei       = (const int*)d_in[4];
  const int*   et       = (const int*)d_in[5];
  const float* W_des = (const float*)d_in[6];
  const float* b_des = (const float*)d_in[7];
  const float* W_tw  = (const float*)d_in[8];
  const float* b_tw  = (const float*)d_in[9];
  const float* W_np  = (const float*)d_in[10];
  const float* b_np  = (const float*)d_in[11];
  const float* W_cp  = (const float*)d_in[12];
  const float* b_cp  = (const float*)d_in[13];
  const float* W_in  = (const float*)d_in[14];
  const float* b_in  = (const float*)d_in[15];
  const float* rel_w  = (const float*)d_in[16];
  const float* root_w = (const float*)d_in[17];
  const float* rgcn_b = (const float*)d_in[18];
  const float* W_o1  = (const float*)d_in[19];
  const float* b_o1  = (const float*)d_in[20];
  const float* W_o2  = (const float*)d_in[21];
  const float* b_o2  = (const float*)d_in[22];

  const int N = in_sizes[0] / 768;   // 50000
  const int E = in_sizes[5];         // 800000

  // workspace layout (floats)
  float* xA  = (float*)d_ws;
  float* xB  = xA + (size_t)N * DD;
  float* h   = xB + (size_t)N * DD;                 // [R, N, 128]
  float* inv = h + (size_t)RR * N * DD;             // [N, R]
  float* Wt  = inv + (size_t)N * RR;                // 8 transposed 128x128 matrices
  // Wt slots: 0..4 = rel_w[r], 5 = root_w, 6 = W_in, 7 = W_o1
  const size_t WSZ = (size_t)DD * DD;

  // one-time weight transposes (recomputed each call; deterministic)
  for (int r = 0; r < RR; ++r)
    transpose128_kernel<<<64, 256, 0, stream>>>(rel_w + (size_t)r * WSZ, Wt + (size_t)r * WSZ);
  transpose128_kernel<<<64, 256, 0, stream>>>(root_w, Wt + 5 * WSZ);
  transpose128_kernel<<<64, 256, 0, stream>>>(W_in,   Wt + 6 * WSZ);
  transpose128_kernel<<<64, 256, 0, stream>>>(W_o1,   Wt + 7 * WSZ);

  // per-(node,relation) mean normalization factors
  zero_kernel<<<(N * RR + 255) / 256, 256, 0, stream>>>(inv, N * RR);
  count_kernel<<<(E + 255) / 256, 256, 0, stream>>>(ei, et, inv, E);
  invert_kernel<<<(N * RR + 255) / 256, 256, 0, stream>>>(inv, N * RR);

  // feature projections -> x0 in xB
  const int ntiles16 = N / 16;                      // 3125
  const dim3 pgrid((ntiles16 + 7) / 8);
  proj_kernel<<<pgrid, 256, 0, stream>>>(des,      768, W_des, b_des, xB, 0,  N);
  proj_kernel<<<pgrid, 256, 0, stream>>>(tweet,    768, W_tw,  b_tw,  xB, 32, N);
  proj_kernel<<<pgrid, 256, 0, stream>>>(num_prop, 6,   W_np,  b_np,  xB, 64, N);
  proj_kernel<<<pgrid, 256, 0, stream>>>(cat_prop, 11,  W_cp,  b_cp,  xB, 96, N);

  const int gblocks = (N + 31) / 32;                // 1563

  // x = lrelu(x0 @ W_in + b_in): xB -> xA
  gemm128_kernel<<<gblocks, 128, 0, stream>>>(xB, Wt + 6 * WSZ, b_in, xA, N, 1);

  float* cur = xA;
  float* nxt = xB;
  for (int l = 0; l < 4; ++l) {
    for (int r = 0; r < RR; ++r)
      gemm128_kernel<<<gblocks, 128, 0, stream>>>(cur, Wt + (size_t)r * WSZ,
                                                  nullptr, h + (size_t)r * N * DD, N, 0);
    // root contribution initializes next x
    gemm128_kernel<<<gblocks, 128, 0, stream>>>(cur, Wt + 5 * WSZ, rgcn_b, nxt, N, 0);
    // mean-aggregated relational messages
    scatter_kernel<<<(E + 7) / 8, 256, 0, stream>>>(ei, et, h, inv, nxt, E, N);
    float* tmp = cur; cur = nxt; nxt = tmp;
  }

  // x = lrelu(x @ W_o1 + b_o1): cur -> nxt
  gemm128_kernel<<<gblocks, 128, 0, stream>>>(cur, Wt + 7 * WSZ, b_o1, nxt, N, 1);
  // out = x @ W_o2 + b_o2
  out_kernel<<<(N + 255) / 256, 256, 0, stream>>>(nxt, W_o2, b_o2, (float*)d_out, N);
}